// RevealModel_26422638805467
// MI455X (gfx1250) — compile-verified
//
#include <hip/hip_runtime.h>

typedef __attribute__((ext_vector_type(2))) float v2f;
typedef __attribute__((ext_vector_type(8))) float v8f;

#define N_NODES 50000
#define N_EDGES 400000
#define IN_DIM 100
#define OUT_DIM 200
#define HID 400
#define NUM_STEPS 6
#define GATE3 600            // 3 * OUT_DIM
#define GPAD 608             // 38 * 16, padded gate width
#define NPAD 208             // 13 * 16, padded N for the 200-col GEMM

static __device__ __forceinline__ v8f wmma_f32(v2f a, v2f b, v8f c) {
    // D = A(16x4 f32) * B(4x16 f32) + C(16x16 f32), exact fp32 WMMA
    return __builtin_amdgcn_wmma_f32_16x16x4_f32(false, a, false, b, (short)0, c,
                                                 false, false);
}

// ---------------------------------------------------------------------------
// h[r, c] = c < 100 ? x[r, c] : 0
__global__ void k_pad(const float* __restrict__ x, float* __restrict__ h) {
    long i = (long)blockIdx.x * 256 + threadIdx.x;
    if (i >= (long)N_NODES * OUT_DIM) return;
    int r = (int)(i / OUT_DIM);
    int c = (int)(i % OUT_DIM);
    h[i] = (c < IN_DIM) ? x[(size_t)r * IN_DIM + c] : 0.0f;
}

__global__ void k_zero(float* __restrict__ p, long n) {
    long i = (long)blockIdx.x * 256 + threadIdx.x;
    if (i < n) p[i] = 0.0f;
}

// WpT[n][k] = n < 200 ? W[k][n] : 0     (transposed + column-padded, 208x200)
__global__ void k_padWT(const float* __restrict__ W, float* __restrict__ WT) {
    int i = blockIdx.x * 256 + threadIdx.x;
    if (i >= NPAD * OUT_DIM) return;
    int n = i / OUT_DIM;
    int k = i % OUT_DIM;
    WT[i] = (n < OUT_DIM) ? W[(size_t)k * OUT_DIM + n] : 0.0f;
}

// wp = gate weight rows padded 600 -> 608 with zeros (row-major flat copy)
__global__ void k_padG(const float* __restrict__ w, float* __restrict__ wp) {
    int i = blockIdx.x * 256 + threadIdx.x;
    if (i >= GPAD * OUT_DIM) return;
    wp[i] = (i < GATE3 * OUT_DIM) ? w[i] : 0.0f;
}

// ---------------------------------------------------------------------------
// m = h @ W : A[50000,200] row-major, BT = WpT[208,200] (B transposed, padded).
// 256 threads = 8 waves; each wave owns 16 rows x 13 N-tiles.
// Inner loop: two unconditional global_load_b64 + one fp32 WMMA.
__global__ void k_gemm_m(const float* __restrict__ A, const float* __restrict__ BT,
                         float* __restrict__ C) {
    const int lane = threadIdx.x & 31;
    const int wave = threadIdx.x >> 5;
    const int half = lane >> 4;      // K sub-slot (0 -> k,k+1 ; 1 -> k+2,k+3)
    const int l15  = lane & 15;
    const int rowBase = (blockIdx.x * 8 + wave) * 16;   // uniform per wave
    if (rowBase >= N_NODES) return;                     // whole wave exits
    const float* Arow = A + (size_t)(rowBase + l15) * OUT_DIM + half * 2;

    for (int tn = 0; tn < 13; ++tn) {
        const int ncol = tn * 16 + l15;                         // 0..207
        const float* Brow = BT + (size_t)ncol * OUT_DIM + half * 2;  // always valid
        v8f acc = {0.f, 0.f, 0.f, 0.f, 0.f, 0.f, 0.f, 0.f};
#pragma unroll 5
        for (int k = 0; k < OUT_DIM; k += 4) {
            v2f a = *(const v2f*)(Arow + k);   // 8B-aligned: row*200+half*2+k even
            v2f b = *(const v2f*)(Brow + k);
            acc = wmma_f32(a, b, acc);
        }
        if (ncol < OUT_DIM) {
            const int rbase = rowBase + half * 8;   // rows exact (50000 % 16 == 0)
#pragma unroll
            for (int j = 0; j < 8; ++j)
                C[(size_t)(rbase + j) * OUT_DIM + ncol] = acc[j];
        }
    }
}

// ---------------------------------------------------------------------------
// agg[dst[e], f] += m[src[e], f]   (agg pre-zeroed; atomics resolve in L2)
__global__ void k_scatter(const float* __restrict__ m, const int* __restrict__ src,
                          const int* __restrict__ dst, float* __restrict__ agg) {
    long i = (long)blockIdx.x * 256 + threadIdx.x;
    if (i >= (long)N_EDGES * OUT_DIM) return;
    int e = (int)(i / OUT_DIM);
    int f = (int)(i % OUT_DIM);
    int s = src[e], d = dst[e];
    atomicAdd(&agg[(size_t)d * OUT_DIM + f], m[(size_t)s * OUT_DIM + f]);
}

// ---------------------------------------------------------------------------
// Fused gate GEMMs + GRU update (in place on h).
// Block = 256 threads (8 waves) handles 16 rows. gi = agg@wih^T, gh = h@whh^T
// computed tile-by-tile with fp32 WMMA into LDS (never touching HBM); weights
// come from the zero-padded wihP/whhP so all loads are unconditional b64.
__global__ void k_gru(const float* __restrict__ agg, float* __restrict__ h,
                      const float* __restrict__ wihP, const float* __restrict__ whhP,
                      const float* __restrict__ bih, const float* __restrict__ bhh) {
    __shared__ float gi[16 * GPAD];
    __shared__ float gh[16 * GPAD];
    const int lane = threadIdx.x & 31;
    const int wave = threadIdx.x >> 5;
    const int half = lane >> 4;
    const int l15  = lane & 15;
    const int rowBase = blockIdx.x * 16;

    const float* Ai = agg + (size_t)(rowBase + l15) * OUT_DIM + half * 2;
    const float* Ah = h   + (size_t)(rowBase + l15) * OUT_DIM + half * 2;

    for (int tn = wave; tn < GPAD / 16; tn += 8) {       // 38 N-tiles over 8 waves
        const int n = tn * 16 + l15;                     // 0..607, rows padded
        const float* Bi = wihP + (size_t)n * OUT_DIM + half * 2;
        const float* Bh = whhP + (size_t)n * OUT_DIM + half * 2;
        v8f ai_acc = {0.f, 0.f, 0.f, 0.f, 0.f, 0.f, 0.f, 0.f};
        v8f ah_acc = {0.f, 0.f, 0.f, 0.f, 0.f, 0.f, 0.f, 0.f};
#pragma unroll 5
        for (int k = 0; k < OUT_DIM; k += 4) {
            v2f a0 = *(const v2f*)(Ai + k);
            v2f b0 = *(const v2f*)(Bi + k);
            v2f a1 = *(const v2f*)(Ah + k);
            v2f b1 = *(const v2f*)(Bh + k);
            ai_acc = wmma_f32(a0, b0, ai_acc);
            ah_acc = wmma_f32(a1, b1, ah_acc);
        }
#pragma unroll
        for (int j = 0; j < 8; ++j) {
            gi[(half * 8 + j) * GPAD + n] = ai_acc[j];
            gh[(half * 8 + j) * GPAD + n] = ah_acc[j];
        }
    }
    __syncthreads();

    for (int idx = threadIdx.x; idx < 16 * OUT_DIM; idx += 256) {
        const int r = idx / OUT_DIM;
        const int c = idx % OUT_DIM;
        const float ir = gi[r * GPAD + c]                + bih[c];
        const float iz = gi[r * GPAD + c + OUT_DIM]      + bih[c + OUT_DIM];
        const float in_ = gi[r * GPAD + c + 2 * OUT_DIM] + bih[c + 2 * OUT_DIM];
        const float hr = gh[r * GPAD + c]                + bhh[c];
        const float hz = gh[r * GPAD + c + OUT_DIM]      + bhh[c + OUT_DIM];
        const float hn = gh[r * GPAD + c + 2 * OUT_DIM]  + bhh[c + 2 * OUT_DIM];
        const float rg = 1.0f / (1.0f + __expf(-(ir + hr)));
        const float zg = 1.0f / (1.0f + __expf(-(iz + hz)));
        const float ng = tanhf(in_ + rg * hn);
        const size_t o = (size_t)(rowBase + r) * OUT_DIM + c;
        h[o] = (1.0f - zg) * ng + zg * h[o];             // only this block's rows
    }
}

// ---------------------------------------------------------------------------
// pooled[c] = sum_r relu(h[r,c])   (pooled pre-zeroed)
__global__ void k_pool(const float* __restrict__ h, float* __restrict__ pooled) {
    const int c = threadIdx.x;
    if (c >= OUT_DIM) return;
    float acc = 0.0f;
    for (int r = blockIdx.x; r < N_NODES; r += gridDim.x) {
        const float v = h[(size_t)r * OUT_DIM + c];
        acc += v > 0.0f ? v : 0.0f;
    }
    atomicAdd(&pooled[c], acc);
}

// ---------------------------------------------------------------------------
// tiny MLP head: 200 -> 400 -> 200 -> 400 -> 2 (+softmax), one block.
__global__ void k_mlp(const float* __restrict__ pooled,
                      const float* __restrict__ ef1w, const float* __restrict__ ef1b,
                      const float* __restrict__ ef2w, const float* __restrict__ ef2b,
                      const float* __restrict__ ef3w, const float* __restrict__ ef3b,
                      const float* __restrict__ clsw, const float* __restrict__ clsb,
                      float* __restrict__ out) {
    __shared__ float p[OUT_DIM], f1[HID], f2[OUT_DIM], f3[HID];
    const int t = threadIdx.x;
    for (int i = t; i < OUT_DIM; i += 512) p[i] = pooled[i];
    __syncthreads();
    for (int j = t; j < HID; j += 512) {
        float s = ef1b[j];
        for (int k = 0; k < OUT_DIM; ++k) s += p[k] * ef1w[(size_t)j * OUT_DIM + k];
        f1[j] = s > 0.f ? s : 0.f;
    }
    __syncthreads();
    for (int j = t; j < OUT_DIM; j += 512) {
        float s = ef2b[j];
        for (int k = 0; k < HID; ++k) s += f1[k] * ef2w[(size_t)j * HID + k];
        f2[j] = s > 0.f ? s : 0.f;
    }
    __syncthreads();
    for (int j = t; j < HID; j += 512) {
        float s = ef3b[j];
        for (int k = 0; k < OUT_DIM; ++k) s += f2[k] * ef3w[(size_t)j * OUT_DIM + k];
        f3[j] = s > 0.f ? s : 0.f;
    }
    __syncthreads();
    if (t == 0) {
        float l0 = clsb[0], l1 = clsb[1];
        for (int k = 0; k < HID; ++k) {
            l0 += f3[k] * clsw[k];
            l1 += f3[k] * clsw[HID + k];
        }
        const float mx = fmaxf(l0, l1);
        const float e0 = __expf(l0 - mx), e1 = __expf(l1 - mx);
        out[0] = e0 / (e0 + e1);
        out[1] = e1 / (e0 + e1);
    }
}

// second tuple output: x passed through, at d_out + 2
__global__ void k_copyx(const float* __restrict__ x, float* __restrict__ out) {
    long i = (long)blockIdx.x * 256 + threadIdx.x;
    if (i < (long)N_NODES * IN_DIM) out[2 + i] = x[i];
}

// ---------------------------------------------------------------------------
extern "C" void kernel_launch(void* const* d_in, const int* in_sizes, int n_in,
                              void* d_out, int out_size, void* d_ws, size_t ws_size,
                              hipStream_t stream) {
    const float* x      = (const float*)d_in[0];
    const int*   eidx   = (const int*)d_in[1];       // [2, 400000]: row0=src, row1=dst
    const float* ggnn_w = (const float*)d_in[2];     // [6, 200, 200]
    const float* wih    = (const float*)d_in[3];     // [600, 200]
    const float* whh    = (const float*)d_in[4];     // [600, 200]
    const float* bih    = (const float*)d_in[5];
    const float* bhh    = (const float*)d_in[6];
    const float* ef1w   = (const float*)d_in[7];
    const float* ef1b   = (const float*)d_in[8];
    const float* ef2w   = (const float*)d_in[9];
    const float* ef2b   = (const float*)d_in[10];
    const float* ef3w   = (const float*)d_in[11];
    const float* ef3b   = (const float*)d_in[12];
    const float* clsw   = (const float*)d_in[13];
    const float* clsb   = (const float*)d_in[14];
    float* out = (float*)d_out;

    const int* src = eidx;
    const int* dst = eidx + N_EDGES;

    // workspace layout (all re-initialized each call)
    const size_t HN = (size_t)N_NODES * OUT_DIM;     // 10,000,000 floats
    float* h      = (float*)d_ws;                    // 40 MB
    float* agg    = h + HN;                          // 40 MB
    float* m      = agg + HN;                        // 40 MB
    float* WpT    = m + HN;                          // 208*200 floats
    float* wihP   = WpT + (size_t)NPAD * OUT_DIM;    // 608*200 floats
    float* whhP   = wihP + (size_t)GPAD * OUT_DIM;   // 608*200 floats
    float* pooled = whhP + (size_t)GPAD * OUT_DIM;   // 200 floats

    const int B = 256;
    const int gHN   = (int)((HN + B - 1) / B);                         // 39063
    const int gEF   = (int)(((long)N_EDGES * OUT_DIM + B - 1) / B);    // 312500
    const int gGemm = (N_NODES / 16 + 7) / 8;                          // 391
    const int gGru  = N_NODES / 16;                                    // 3125
    const int gX    = (int)(((long)N_NODES * IN_DIM + B - 1) / B);     // 19532
    const int gWT   = (NPAD * OUT_DIM + B - 1) / B;                    // 163
    const int gGP   = (GPAD * OUT_DIM + B - 1) / B;                    // 475

    k_pad<<<gHN, B, 0, stream>>>(x, h);
    k_padG<<<gGP, B, 0, stream>>>(wih, wihP);
    k_padG<<<gGP, B, 0, stream>>>(whh, whhP);

    for (int step = 0; step < NUM_STEPS; ++step) {
        const float* W = ggnn_w + (size_t)step * OUT_DIM * OUT_DIM;
        k_padWT<<<gWT, B, 0, stream>>>(W, WpT);
        k_gemm_m<<<gGemm, B, 0, stream>>>(h, WpT, m);
        k_zero<<<gHN, B, 0, stream>>>(agg, (long)HN);
        k_scatter<<<gEF, B, 0, stream>>>(m, src, dst, agg);
        k_gru<<<gGru, B, 0, stream>>>(agg, h, wihP, whhP, bih, bhh);
    }

    k_zero<<<1, B, 0, stream>>>(pooled, OUT_DIM);
    k_pool<<<128, B, 0, stream>>>(h, pooled);
    k_mlp<<<1, 512, 0, stream>>>(pooled, ef1w, ef1b, ef2w, ef2b, ef3w, ef3b,
                                 clsw, clsb, out);
    k_copyx<<<gX, B, 0, stream>>>(x, out);
}